// Bert_53884659695737
// MI455X (gfx1250) — compile-verified
//
#include <hip/hip_runtime.h>
#include <stdint.h>

typedef __attribute__((ext_vector_type(2))) float v2f;
typedef __attribute__((ext_vector_type(8))) float v8f;

#define NB 64
#define NS 512
#define NW 400
#define ND 1024
#define WT 16           // words per block tile
#define NWT (NW / WT)   // 25 word tiles

// One workgroup per (batch, 16-word tile). 256 threads = 8 wave32.
// Wave j owns d-columns [128j, 128j+128) as 8 WMMA n-tiles of 16.
// K-loop over the bracket [st[w0], ed[w0+15]) in chunks of 4 rows,
// staged into LDS via async-to-LDS loads (double buffered, per-wave).
__global__ __launch_bounds__(256) void span_mean_wmma(
    const float* __restrict__ bert,   // [B,S,D] f32
    const int*   __restrict__ offs,   // [B,W,2] i32
    const int*   __restrict__ xmask,  // [B,W]   i32
    float* __restrict__ out)          // [B,W,D] f32
{
    __shared__ float s_scale[WT];
    __shared__ int   s_lohi[2];
    __shared__ float s_stage[8 * 2 * 4 * 128]; // 8 waves x 2 bufs x 4 rows x 128 f32 = 32 KB

    const int wt   = blockIdx.x;           // 0..24
    const int b    = blockIdx.y;           // 0..63
    const int w0   = wt * WT;
    const int tid  = threadIdx.x;
    const int wave = tid >> 5;
    const int lane = tid & 31;
    const int n    = lane & 15;            // A-row m / B-col n / C-col n
    const int kb   = (lane >> 4) << 1;     // K base: 0 (lanes 0-15) or 2 (lanes 16-31)

    // Span bounds for this lane's A-matrix row (word w0+n).
    const int2 se   = ((const int2*)offs)[b * NW + w0 + n];
    const int  myst = se.x, myed = se.y;

    // Per-word output scale: valid ? 1/max(len,1) : 0
    if (tid < WT) {
        const int2 q    = ((const int2*)offs)[b * NW + w0 + tid];
        const int  len  = q.y - q.x;
        const bool val  = (xmask[b * NW + w0 + tid] != 0) && (len > 0);
        const float l   = (len < 1) ? 1.0f : (float)len;
        s_scale[tid]    = val ? (1.0f / l) : 0.0f;
    }
    // Bracket: boundaries are sorted per batch row, so [st[w0], ed[w0+15])
    // contains every span of this word tile.
    if (tid == 0)  s_lohi[0] = ((const int2*)offs)[b * NW + w0].x;
    if (tid == 32) s_lohi[1] = ((const int2*)offs)[b * NW + w0 + WT - 1].y;
    __syncthreads();

    const int lo = s_lohi[0];
    const int hi = s_lohi[1];
    const int nk = (hi > lo) ? ((hi - lo + 3) >> 2) : 0;   // block-uniform

    v8f acc[8];
#pragma unroll
    for (int t = 0; t < 8; ++t) acc[t] = (v8f){0.f,0.f,0.f,0.f,0.f,0.f,0.f,0.f};

    // Per-lane staging addresses: lane covers 16B of this wave's 512B row slice.
    const float*   gbase = bert + ((size_t)b * NS) * ND + wave * 128 + lane * 4;
    const uint32_t lds0  = (uint32_t)(uintptr_t)&s_stage[0] + (uint32_t)(wave * 4096);

    auto stage = [&](int chunk, int buf) {
        const int s0s = lo + 4 * chunk;
#pragma unroll
        for (int r = 0; r < 4; ++r) {
            int s = s0s + r;
            s = (s < NS) ? s : (NS - 1);   // clamp; rows past the bracket are masked to 0 by A
            const uint64_t gp = (uint64_t)(uintptr_t)(gbase + (size_t)s * ND);
            const uint32_t lp = lds0 + (uint32_t)((buf * 4 + r) * 512 + lane * 16);
            asm volatile("global_load_async_to_lds_b128 %0, %1, off"
                         :: "v"(lp), "v"(gp) : "memory");
        }
    };

    if (nk > 0) {
        stage(0, 0);
        for (int i = 0; i < nk; ++i) {
            const int buf = i & 1;
            if (i + 1 < nk) {
                stage(i + 1, buf ^ 1);                          // prefetch next chunk
                asm volatile("s_wait_asynccnt 4" ::: "memory"); // current chunk landed
            } else {
                asm volatile("s_wait_asynccnt 0" ::: "memory");
            }
            const int s0 = lo + 4 * i;

            // A: 16x4 f32 0/1 mask (VGPR0 = K0/K2, VGPR1 = K1/K3 across lane halves)
            v2f a;
            a.x = (s0 + kb     >= myst && s0 + kb     < myed) ? 1.0f : 0.0f;
            a.y = (s0 + kb + 1 >= myst && s0 + kb + 1 < myed) ? 1.0f : 0.0f;

            const float* lrow = &s_stage[wave * 1024 + buf * 512];
#pragma unroll
            for (int t = 0; t < 8; ++t) {
                // B: 4x16 f32, K-halves mirroring A's layout
                v2f bb;
                bb.x = lrow[(kb    ) * 128 + t * 16 + n];
                bb.y = lrow[(kb + 1) * 128 + t * 16 + n];
                acc[t] = __builtin_amdgcn_wmma_f32_16x16x4_f32(
                    false, a, false, bb, (short)0, acc[t], false, false);
            }
        }
    }

    // C layout: VGPR r holds rows m=r (lanes 0-15) and m=r+8 (lanes 16-31), col = n.
    const int mo_base = (lane >> 4) << 3;
#pragma unroll
    for (int t = 0; t < 8; ++t) {
        const int dcol = wave * 128 + t * 16 + n;
#pragma unroll
        for (int r = 0; r < 8; ++r) {
            const int mo = mo_base + r;
            out[((size_t)(b * NW + w0 + mo)) * ND + dcol] = acc[t][r] * s_scale[mo];
        }
    }
}

extern "C" void kernel_launch(void* const* d_in, const int* in_sizes, int n_in,
                              void* d_out, int out_size, void* d_ws, size_t ws_size,
                              hipStream_t stream) {
    (void)in_sizes; (void)n_in; (void)out_size; (void)d_ws; (void)ws_size;
    const float* bert  = (const float*)d_in[0];
    const int*   offs  = (const int*)d_in[1];
    const int*   xmask = (const int*)d_in[2];
    float*       out   = (float*)d_out;

    dim3 grid(NWT, NB);   // 25 x 64 workgroups
    span_mean_wmma<<<grid, 256, 0, stream>>>(bert, offs, xmask, out);
}